// Decoder_49297634624194
// MI455X (gfx1250) — compile-verified
//
#include <hip/hip_runtime.h>
#include <hip/hip_bf16.h>
#include <cstddef>

// Problem dims (from reference)
#define BB 64
#define TT 59
#define SS 32
#define HH 512
#define EE 300
#define VV 34004
#define MM 1024
#define KX 1344          // padded (E+H+H)=1324 -> 1344 (multiple of 32)
#define G4H 2048         // 4*H
#define ACW 1536         // H + M

typedef float v2f __attribute__((ext_vector_type(2)));
typedef float v8f __attribute__((ext_vector_type(8)));

__device__ __forceinline__ float sigmoidf_(float x) { return 1.0f / (1.0f + __expf(-x)); }

// ---------------------------------------------------------------------------
// Generic fp32 WMMA GEMM: C[Mr x N] = A[Mr x K] @ B[K x N] (+bias)
// Mr multiple of 64 (grid.y*64), K multiple of 32, N edge handled by
// column-clamped loads (garbage only lands in never-stored lanes' columns).
// Block: 128 threads (4 waves). Block tile 64 x (64*NT). Wave w owns NT
// 16-wide N tiles; A fragment (LDS, padded stride 36, conflict-free) is
// reused across all NT B fragments -> 4*NT WMMAs per 4 ds_load_b64.
// B addressed by a running per-lane pointer (no per-load multiplies, no
// exec-mask churn in the hot loop).
// ---------------------------------------------------------------------------
template <int NT>
__global__ __launch_bounds__(128) void gemm_wmma_f32(
    const float* __restrict__ A, int lda,
    const float* __restrict__ Bm, int ldb,
    float* __restrict__ C, long ldc,
    int K, int N, const float* __restrict__ bias)
{
    __shared__ float As[64 * 36];

    const int tid  = threadIdx.x;
    const int lane = tid & 31;
    const int wv   = tid >> 5;              // 0..3
    const int ml   = lane & 15;             // row-in-tile / col-in-tile
    const int koff = (lane >> 4) << 1;      // 0 or 2 (A/B k sub-offset)

    const int m0 = blockIdx.y * 64;
    const size_t ldbs = (size_t)ldb;

    int  ncol[NT];
    bool inb[NT];
    const float* bp[NT];                    // running B pointers (advance 4*ldb/kk)
#pragma unroll
    for (int nt = 0; nt < NT; ++nt) {
        ncol[nt] = blockIdx.x * (64 * NT) + wv * (16 * NT) + nt * 16 + ml;
        inb[nt] = (ncol[nt] < N);
        int ncl = inb[nt] ? ncol[nt] : (N - 1);   // clamp: always-valid loads
        bp[nt] = Bm + (size_t)koff * ldbs + ncl;
    }

    v8f acc[4 * NT];
#pragma unroll
    for (int i = 0; i < 4 * NT; ++i)
#pragma unroll
        for (int r = 0; r < 8; ++r) acc[i][r] = 0.0f;

    for (int k0 = 0; k0 < K; k0 += 32) {
        __syncthreads();
        {   // stage A[64 x 32] chunk into LDS, float4 loads/stores
            const int row = tid >> 1;
            const int seg = (tid & 1) * 16;
            const float4* gp =
                reinterpret_cast<const float4*>(A + (size_t)(m0 + row) * lda + k0 + seg);
#pragma unroll
            for (int i = 0; i < 4; ++i) {
                float4 vv = gp[i];
                *reinterpret_cast<float4*>(&As[row * 36 + seg + 4 * i]) = vv;
            }
        }
        __syncthreads();

#pragma unroll
        for (int kk = 0; kk < 32; kk += 4) {
            v2f bf_[NT];
#pragma unroll
            for (int nt = 0; nt < NT; ++nt) {
                bf_[nt].x = bp[nt][0];
                bf_[nt].y = bp[nt][ldbs];
                bp[nt] += 4 * ldbs;
            }
#pragma unroll
            for (int mt = 0; mt < 4; ++mt) {
                v2f af;
                af.x = As[(mt * 16 + ml) * 36 + kk + koff];
                af.y = As[(mt * 16 + ml) * 36 + kk + koff + 1];
#pragma unroll
                for (int nt = 0; nt < NT; ++nt) {
                    acc[mt * NT + nt] = __builtin_amdgcn_wmma_f32_16x16x4_f32(
                        false, af, false, bf_[nt], (short)0, acc[mt * NT + nt],
                        false, false);
                }
            }
        }
    }

    const int rbase = (lane >> 4) * 8;   // lanes 0-15: rows r, lanes 16-31: rows r+8
#pragma unroll
    for (int nt = 0; nt < NT; ++nt) {
        if (inb[nt]) {
            float bval = (bias != nullptr) ? bias[ncol[nt]] : 0.0f;
#pragma unroll
            for (int mt = 0; mt < 4; ++mt) {
#pragma unroll
                for (int r = 0; r < 8; ++r) {
                    const long grow = m0 + mt * 16 + r + rbase;
                    C[(size_t)grow * ldc + ncol[nt]] = acc[mt * NT + nt][r] + bval;
                }
            }
        }
    }
}

// ---------------------------------------------------------------------------
// One-time: pack Wcat[KX x 4H] = [Wx ; Uh ; zeros]
// ---------------------------------------------------------------------------
__global__ void pack_W_kernel(const float* __restrict__ Wx, const float* __restrict__ Uh,
                              float* __restrict__ Wcat)
{
    int idx = blockIdx.x * blockDim.x + threadIdx.x;
    const int total = KX * G4H;
    if (idx >= total) return;
    int k = idx / G4H, n = idx % G4H;
    float v = 0.0f;
    if (k < EE + HH)            v = Wx[k * G4H + n];
    else if (k < EE + 2 * HH)   v = Uh[(k - (EE + HH)) * G4H + n];
    Wcat[idx] = v;
}

// One-time: h=enc_h, c=enc_c, attn=0
__global__ void init_state_kernel(const float* __restrict__ eh, const float* __restrict__ ec,
                                  float* __restrict__ h, float* __restrict__ c,
                                  float* __restrict__ attn)
{
    int idx = blockIdx.x * blockDim.x + threadIdx.x;
    if (idx >= BB * HH) return;
    h[idx] = eh[idx];
    c[idx] = ec[idx];
    attn[idx] = 0.0f;
}

// Per step: xcat[b] = [embedding[dec_input[b,t]] | attn[b] | h[b] | 0pad]
__global__ void pack_x_kernel(const int* __restrict__ dec_input,
                              const float* __restrict__ emb,
                              const float* __restrict__ attn,
                              const float* __restrict__ h,
                              float* __restrict__ xcat, int t)
{
    int idx = blockIdx.x * blockDim.x + threadIdx.x;
    if (idx >= BB * KX) return;
    int b = idx / KX, j = idx % KX;
    float v = 0.0f;
    if (j < EE) {
        int tok = dec_input[b * TT + t];
        v = emb[(size_t)tok * EE + j];
    } else if (j < EE + HH) {
        v = attn[b * HH + (j - EE)];
    } else if (j < EE + 2 * HH) {
        v = h[b * HH + (j - (EE + HH))];
    }
    xcat[idx] = v;
}

// Per step: LSTM pointwise cell. z[b, 4H] -> h,c updated; h also into ac[:, :H]
__global__ void lstm_cell_kernel(const float* __restrict__ z,
                                 float* __restrict__ h, float* __restrict__ c,
                                 float* __restrict__ ac)
{
    int idx = blockIdx.x * blockDim.x + threadIdx.x;
    if (idx >= BB * HH) return;
    int b = idx / HH, j = idx % HH;
    const float* zr = z + (size_t)b * G4H;
    float zi = zr[j], zf = zr[HH + j], zg = zr[2 * HH + j], zo = zr[3 * HH + j];
    float c2 = sigmoidf_(zf) * c[idx] + sigmoidf_(zi) * tanhf(zg);
    float h2 = sigmoidf_(zo) * tanhf(c2);
    c[idx] = c2;
    h[idx] = h2;
    ac[(size_t)b * ACW + j] = h2;      // first half of [h2 | ctx]
}

// Per step: Bahdanau attention. One block per batch row, 256 threads (8 waves).
// scores_s = v . tanh(keys[b,s,:] + q[b,:]); softmax over S; ctx -> ac[:, H:]
__global__ __launch_bounds__(256) void attention_kernel(
    const float* __restrict__ keys, const float* __restrict__ q,
    const float* __restrict__ vvec, const float* __restrict__ enc,
    float* __restrict__ ac)
{
    __shared__ float sc[SS];
    __shared__ float al[SS];
    const int b = blockIdx.x;
    const int tid = threadIdx.x;
    const int lane = tid & 31;
    const int wv = tid >> 5;

    // scores: wave wv handles s = wv*4 .. wv*4+3
#pragma unroll
    for (int i = 0; i < 4; ++i) {
        int s = wv * 4 + i;
        const float* kr = keys + ((size_t)(b * SS + s)) * HH;
        const float* qr = q + (size_t)b * HH;
        float accv = 0.0f;
        for (int hh = lane; hh < HH; hh += 32)
            accv += vvec[hh] * tanhf(kr[hh] + qr[hh]);
#pragma unroll
        for (int off = 16; off > 0; off >>= 1)
            accv += __shfl_down(accv, off, 32);
        if (lane == 0) sc[s] = accv;
    }
    __syncthreads();

    // softmax over 32 scores (wave 0, all-lanes reduce via xor shuffles)
    if (wv == 0) {
        float val = sc[lane];
        float mx = val;
#pragma unroll
        for (int off = 16; off > 0; off >>= 1)
            mx = fmaxf(mx, __shfl_xor(mx, off, 32));
        float e = __expf(val - mx);
        float sum = e;
#pragma unroll
        for (int off = 16; off > 0; off >>= 1)
            sum += __shfl_xor(sum, off, 32);
        al[lane] = e / sum;
    }
    __syncthreads();

    // ctx[b, m] = sum_s al[s] * enc[b, s, m] -> ac[b, H + m]
#pragma unroll
    for (int ii = 0; ii < MM / 256; ++ii) {
        int m = tid + 256 * ii;
        float accv = 0.0f;
#pragma unroll 4
        for (int s = 0; s < SS; ++s)
            accv += al[s] * enc[((size_t)(b * SS + s)) * MM + m];
        ac[(size_t)b * ACW + HH + m] = accv;
    }
}

// ---------------------------------------------------------------------------
extern "C" void kernel_launch(void* const* d_in, const int* in_sizes, int n_in,
                              void* d_out, int out_size, void* d_ws, size_t ws_size,
                              hipStream_t stream)
{
    const int*   dec_input = (const int*)  d_in[0];
    const float* enc_out   = (const float*)d_in[1];   // [B,S,M]
    const float* enc_h     = (const float*)d_in[2];
    const float* enc_c     = (const float*)d_in[3];
    const float* embedding = (const float*)d_in[4];   // [V,E]
    const float* Wx        = (const float*)d_in[5];   // [E+H,4H]
    const float* Uh        = (const float*)d_in[6];   // [H,4H]
    const float* bgate     = (const float*)d_in[7];   // [4H]
    const float* Wm        = (const float*)d_in[8];   // [M,H]
    const float* Wq        = (const float*)d_in[9];   // [H,H]
    const float* vvec      = (const float*)d_in[10];  // [H]
    const float* Wa        = (const float*)d_in[11];  // [H+M,H]
    const float* Wf        = (const float*)d_in[12];  // [H,V]
    const float* bf        = (const float*)d_in[13];  // [V]
    float* out = (float*)d_out;                        // [B,T,V]

    // Workspace layout (floats)
    float* ws = (float*)d_ws;
    size_t off = 0;
    float* keys = ws + off; off += (size_t)BB * SS * HH;   // 1,048,576
    float* Wcat = ws + off; off += (size_t)KX * G4H;       // 2,752,512
    float* zbuf = ws + off; off += (size_t)BB * G4H;       // 131,072
    float* xcat = ws + off; off += (size_t)BB * KX;        // 86,016
    float* hbuf = ws + off; off += (size_t)BB * HH;
    float* cbuf = ws + off; off += (size_t)BB * HH;
    float* attn = ws + off; off += (size_t)BB * HH;
    float* qbuf = ws + off; off += (size_t)BB * HH;
    float* acb  = ws + off; off += (size_t)BB * ACW;       // total ~4.25M floats

    // ---- one-time setup ----
    pack_W_kernel<<<(KX * G4H + 255) / 256, 256, 0, stream>>>(Wx, Uh, Wcat);
    init_state_kernel<<<(BB * HH + 255) / 256, 256, 0, stream>>>(enc_h, enc_c,
                                                                 hbuf, cbuf, attn);
    // keys[B*S, H] = enc_out[B*S, M] @ Wm[M, H]
    gemm_wmma_f32<1><<<dim3(HH / 64, (BB * SS) / 64), 128, 0, stream>>>(
        enc_out, MM, Wm, HH, keys, HH, MM, HH, nullptr);

    // ---- recurrent decode ----
    for (int t = 0; t < TT; ++t) {
        pack_x_kernel<<<(BB * KX + 255) / 256, 256, 0, stream>>>(
            dec_input, embedding, attn, hbuf, xcat, t);
        // z = [x|attn|h] @ [Wx;Uh] + b   (64 x 2048, K=1344)
        gemm_wmma_f32<1><<<dim3(G4H / 64, BB / 64), 128, 0, stream>>>(
            xcat, KX, Wcat, G4H, zbuf, G4H, KX, G4H, bgate);
        lstm_cell_kernel<<<(BB * HH + 255) / 256, 256, 0, stream>>>(
            zbuf, hbuf, cbuf, acb);
        // q = h2 @ Wq   (64 x 512, K=512)
        gemm_wmma_f32<1><<<dim3(HH / 64, BB / 64), 128, 0, stream>>>(
            hbuf, HH, Wq, HH, qbuf, HH, HH, HH, nullptr);
        attention_kernel<<<BB, 256, 0, stream>>>(keys, qbuf, vvec, enc_out, acb);
        // attn2 = [h2|ctx] @ Wa  (64 x 512, K=1536) -> attn buffer
        gemm_wmma_f32<1><<<dim3(HH / 64, BB / 64), 128, 0, stream>>>(
            acb, ACW, Wa, HH, attn, HH, ACW, HH, nullptr);
        // logits = attn2 @ Wf + bf  (64 x 34004, K=512), NT=2 wide tiles
        gemm_wmma_f32<2><<<dim3((VV + 127) / 128, BB / 64), 128, 0, stream>>>(
            attn, HH, Wf, VV, out + (size_t)t * VV, (long)TT * VV, HH, VV, bf);
    }
}